// SAGE_25151328485677
// MI455X (gfx1250) — compile-verified
//
#include <hip/hip_runtime.h>

#define IN_DIM  128
#define HID_DIM 256
#define OUT_DIM 47

typedef __attribute__((ext_vector_type(2))) float v2f;
typedef __attribute__((ext_vector_type(4))) float v4f;
typedef __attribute__((ext_vector_type(8))) float v8f;

// Hardware FP32 atomic add (global_atomic_add_f32); accumulator is L2-resident
// (192MB L2 >> 102MB table) so these stay on-chip.
static __device__ __forceinline__ void atomAddF(float* p, float v) {
    unsafeAtomicAdd(p, v);
}

// ---------------------------------------------------------------- utility ---
__global__ void fill_kernel(float* __restrict__ p, float v, size_t n) {
    size_t i = (size_t)blockIdx.x * blockDim.x + threadIdx.x;
    size_t stride = (size_t)gridDim.x * blockDim.x;
    for (; i < n; i += stride) p[i] = v;
}

__global__ void degree_kernel(const int* __restrict__ dst, float* __restrict__ deg, int E) {
    int e = blockIdx.x * blockDim.x + threadIdx.x;
    if (e < E) atomAddF(&deg[dst[e]], 1.0f);
}

__global__ void inv_kernel(float* __restrict__ d, int n) {
    int i = blockIdx.x * blockDim.x + threadIdx.x;
    if (i < n) d[i] = 1.0f / fmaxf(d[i], 1.0f);
}

// --------------------------------------------------------------- scatter ----
// Edge-parallel scatter-add, 4 features per thread (d % 4 == 0).
__global__ void scatter_add_vec4(const float* __restrict__ X, float* __restrict__ ACC,
                                 const int* __restrict__ src, const int* __restrict__ dst,
                                 int E, int d) {
    const int per = d >> 2;
    long long tid = (long long)blockIdx.x * blockDim.x + threadIdx.x;
    long long total = (long long)E * per;
    if (tid >= total) return;
    int e = (int)(tid / per);
    int c = ((int)(tid % per)) << 2;
    const v4f v = *(const v4f*)(X + (size_t)src[e] * d + c);
    float* p = ACC + (size_t)dst[e] * d + c;
    atomAddF(p + 0, v.x);
    atomAddF(p + 1, v.y);
    atomAddF(p + 2, v.z);
    atomAddF(p + 3, v.w);
}

// Scalar variant for d = 47 (layer-3, transform-first trick).
__global__ void scatter_add_scalar(const float* __restrict__ X, float* __restrict__ ACC,
                                   const int* __restrict__ src, const int* __restrict__ dst,
                                   int E, int d) {
    long long tid = (long long)blockIdx.x * blockDim.x + threadIdx.x;
    long long total = (long long)E * d;
    if (tid >= total) return;
    int e = (int)(tid / d);
    int c = (int)(tid % d);
    atomAddF(ACC + (size_t)dst[e] * d + c, X[(size_t)src[e] * d + c]);
}

// ------------------------------------------------------------ fused GEMM ----
// Out[i,:] = X@Wself (+ (ACC[i,:]*inv[i])@Wneigh) (+ bias) (+ addsrc[i,:]*inv[i]); optional ReLU.
//
// Grid: blockIdx.y = 64-column strip, blockIdx.x = group of 8 row-tiles.
// The block stages the (transposed, zero-padded) weight strip(s) in LDS once;
// all 8 waves consume it via conflict-free ds_load_b64 (stride K+4). Columns
// >= Nout are zero-filled in LDS, so the WMMA loop needs no guards at all.
//
// One wave computes a 16x64 output strip = 4 x V_WMMA_F32_16X16X4_F32 tiles.
// A 16x4 layout: lane(0..15)=row M, K={0,1} ; lane(16..31)=row M, K={2,3}.
// B 4x16 layout: lane(0..15)=col N, K={0,1} ; lane(16..31)=col N, K={2,3}.
// C/D 16x16  : VGPR j -> M = j + 8*(lane>=16), N = lane&15.
template <bool HAS_NEIGH, bool HAS_ADD, bool RELU>
__launch_bounds__(256)
__global__ void sage_gemm_wmma(const float* __restrict__ X,
                               const float* __restrict__ ACC,
                               const float* __restrict__ inv,
                               const float* __restrict__ Wself,
                               const float* __restrict__ Wneigh,
                               const float* __restrict__ bias,
                               const float* __restrict__ addsrc,
                               float* __restrict__ Out,
                               int M, int K, int Nout) {
    extern __shared__ float lds_mem[];
    const int Kp = K + 4;                 // pad -> conflict-free b64 B reads
    float* ldsS = lds_mem;
    float* ldsN = lds_mem + 64 * Kp;      // only used when HAS_NEIGH

    const int colBase = blockIdx.y << 6;  // 64 columns per block

    // ---- cooperative staged transpose of the weight strip(s) into LDS ----
    // t -> (col = t&63, k = t>>6): global reads are 64-float contiguous runs,
    // LDS writes land on 32 distinct banks (stride Kp, Kp%64 == 4).
    for (int t = threadIdx.x; t < (K << 6); t += blockDim.x) {
        const int col = t & 63;
        const int k   = t >> 6;
        const int gc  = colBase + col;
        float vs = 0.0f, vn = 0.0f;
        if (gc < Nout) {
            vs = Wself[(size_t)k * Nout + gc];
            if constexpr (HAS_NEIGH) vn = Wneigh[(size_t)k * Nout + gc];
        }
        ldsS[col * Kp + k] = vs;
        if constexpr (HAS_NEIGH) ldsN[col * Kp + k] = vn;
    }
    __syncthreads();

    const int rowTiles = (M + 15) >> 4;
    const int rowTile  = (blockIdx.x << 3) + (threadIdx.x >> 5);
    if (rowTile >= rowTiles) return;      // after the barrier: safe

    const int lane = threadIdx.x & 31;
    const int half = lane >> 4;           // 0 or 1
    const int l16  = lane & 15;
    const int row0 = rowTile << 4;

    const int rowA = min(row0 + l16, M - 1);     // clamp (M%16==0 here anyway)
    const float* xrow = X + (size_t)rowA * K + 2 * half;

    // Per-lane LDS base for each of the 4 sub-tiles' B fragments.
    const float* bS[4];
    const float* bN[4];
#pragma unroll
    for (int n = 0; n < 4; ++n) {
        const int lc = n * 16 + l16;      // local column 0..63
        bS[n] = ldsS + lc * Kp + 2 * half;
        if constexpr (HAS_NEIGH) bN[n] = ldsN + lc * Kp + 2 * half;
    }

    v8f cacc[4] = {};

    if constexpr (HAS_NEIGH) {
        __builtin_prefetch(ACC + (size_t)rowA * K, 0, 1);  // warm the neigh pass
    }

    // ---- self path: X @ Wself ----
#pragma unroll 2
    for (int k = 0; k < K; k += 4) {
        v2f a = *(const v2f*)(xrow + k);
#pragma unroll
        for (int n = 0; n < 4; ++n) {
            v2f b = *(const v2f*)(bS[n] + k);
            cacc[n] = __builtin_amdgcn_wmma_f32_16x16x4_f32(
                false, a, false, b, (short)0, cacc[n], false, false);
        }
    }

    // ---- neighbor path: (ACC * inv) @ Wneigh ----
    if constexpr (HAS_NEIGH) {
        const float  s    = inv[rowA];
        const float* arow = ACC + (size_t)rowA * K + 2 * half;
#pragma unroll 2
        for (int k = 0; k < K; k += 4) {
            v2f a = *(const v2f*)(arow + k);
            a.x *= s; a.y *= s;
#pragma unroll
            for (int n = 0; n < 4; ++n) {
                v2f b = *(const v2f*)(bN[n] + k);
                cacc[n] = __builtin_amdgcn_wmma_f32_16x16x4_f32(
                    false, a, false, b, (short)0, cacc[n], false, false);
            }
        }
    }

    // ---- epilogue: bias, scaled residual add, ReLU, guarded store ----
#pragma unroll
    for (int n = 0; n < 4; ++n) {
        const int col = colBase + n * 16 + l16;
        if (col >= Nout) continue;
        const float bv = bias ? bias[col] : 0.0f;
#pragma unroll
        for (int j = 0; j < 8; ++j) {
            const int row = row0 + half * 8 + j;
            if (row >= M) continue;
            float v = cacc[n][j] + bv;
            if constexpr (HAS_ADD)
                v += addsrc[(size_t)row * Nout + col] * inv[row];
            if constexpr (RELU)
                v = fmaxf(v, 0.0f);
            Out[(size_t)row * Nout + col] = v;
        }
    }
}

// ------------------------------------------------------------------ host ----
static inline int cdiv_ll(long long a, int b) { return (int)((a + b - 1) / b); }

extern "C" void kernel_launch(void* const* d_in, const int* in_sizes, int n_in,
                              void* d_out, int out_size, void* d_ws, size_t ws_size,
                              hipStream_t stream) {
    const float* x   = (const float*)d_in[0];
    const int*   src = (const int*)  d_in[1];
    const int*   dst = (const int*)  d_in[2];
    const float* ws1 = (const float*)d_in[3];   // w_self1  128x256
    const float* wn1 = (const float*)d_in[4];   // w_neigh1 128x256
    const float* b1  = (const float*)d_in[5];
    const float* ws2 = (const float*)d_in[6];   // 256x256
    const float* wn2 = (const float*)d_in[7];
    const float* b2  = (const float*)d_in[8];
    const float* ws3 = (const float*)d_in[9];   // 256x47
    const float* wn3 = (const float*)d_in[10];
    const float* b3  = (const float*)d_in[11];

    const int N = in_sizes[0] / IN_DIM;   // 100000
    const int E = in_sizes[1];            // 1600000

    // Workspace layout (floats): inv | acc(N*256) | h1(N*256, reused for z) | h2(N*256)
    float* inv = (float*)d_ws;
    float* acc = inv + (((size_t)N + 255) & ~(size_t)255);
    float* h1  = acc + (size_t)N * HID_DIM;
    float* h2  = h1  + (size_t)N * HID_DIM;
    float* z   = h1;                      // h1 is dead by the time z is produced

    const int T = 256;
    const int rowBlocks = cdiv_ll(((long long)N + 15) / 16, 8);
    auto smem = [](int K, bool neigh) {
        return (size_t)(neigh ? 2 : 1) * 64 * (K + 4) * sizeof(float);
    };

    // ---- degrees -> inv = 1/max(deg,1) ----
    fill_kernel<<<cdiv_ll(N, T), T, 0, stream>>>(inv, 0.0f, (size_t)N);
    degree_kernel<<<cdiv_ll(E, T), T, 0, stream>>>(dst, inv, E);
    inv_kernel<<<cdiv_ll(N, T), T, 0, stream>>>(inv, N);

    // ---- layer 1: agg(x,128) ; h1 = relu(x@ws1 + (acc*inv)@wn1 + b1) ----
    fill_kernel<<<cdiv_ll((long long)N * IN_DIM, T), T, 0, stream>>>(acc, 0.0f, (size_t)N * IN_DIM);
    scatter_add_vec4<<<cdiv_ll((long long)E * (IN_DIM / 4), T), T, 0, stream>>>(x, acc, src, dst, E, IN_DIM);
    sage_gemm_wmma<true, false, true>
        <<<dim3(rowBlocks, HID_DIM / 64), T, smem(IN_DIM, true), stream>>>(
        x, acc, inv, ws1, wn1, b1, nullptr, h1, N, IN_DIM, HID_DIM);

    // ---- layer 2 ----
    fill_kernel<<<cdiv_ll((long long)N * HID_DIM, T), T, 0, stream>>>(acc, 0.0f, (size_t)N * HID_DIM);
    scatter_add_vec4<<<cdiv_ll((long long)E * (HID_DIM / 4), T), T, 0, stream>>>(h1, acc, src, dst, E, HID_DIM);
    sage_gemm_wmma<true, false, true>
        <<<dim3(rowBlocks, HID_DIM / 64), T, smem(HID_DIM, true), stream>>>(
        h1, acc, inv, ws2, wn2, b2, nullptr, h2, N, HID_DIM, HID_DIM);

    // ---- layer 3 (transform-first: aggregate in 47-dim instead of 256-dim) ----
    // z = h2 @ wn3
    sage_gemm_wmma<false, false, false>
        <<<dim3(rowBlocks, 1), T, smem(HID_DIM, false), stream>>>(
        h2, nullptr, inv, wn3, nullptr, nullptr, nullptr, z, N, HID_DIM, OUT_DIM);
    // acc = segment_sum(z[src] -> dst)
    fill_kernel<<<cdiv_ll((long long)N * OUT_DIM, T), T, 0, stream>>>(acc, 0.0f, (size_t)N * OUT_DIM);
    scatter_add_scalar<<<cdiv_ll((long long)E * OUT_DIM, T), T, 0, stream>>>(z, acc, src, dst, E, OUT_DIM);
    // out = h2 @ ws3 + b3 + acc*inv   (no relu on final layer)
    sage_gemm_wmma<false, true, false>
        <<<dim3(rowBlocks, 1), T, smem(HID_DIM, false), stream>>>(
        h2, nullptr, inv, ws3, nullptr, b3, acc, (float*)d_out, N, HID_DIM, OUT_DIM);
}